// RoIAlign_64149631533379
// MI455X (gfx1250) — compile-verified
//
#include <hip/hip_runtime.h>
#include <hip/hip_bf16.h>

typedef __attribute__((ext_vector_type(2))) float v2f;
typedef __attribute__((ext_vector_type(8))) float v8f;

#define FH 200
#define FW 200
#define NC 256
#define CROPV 14
#define SSCALE 0.25f

// One block (512 thr = 16 waves) per ROI; wave w handles channels [16w, 16w+16).
// Per output row i: D(16c x 16j) = A(16c x 28k) * B(28k x 14j) via 7 chained
// V_WMMA_F32_16X16X4_F32 accumulations. A = y-blended gathered columns,
// B = x-interp weights (2 nonzeros/col). Validity masks folded into A rows /
// B cols so masked outputs are exactly 0.0f.
__global__ __launch_bounds__(512) void roialign_wmma_kernel(
    const float* __restrict__ feat, const float* __restrict__ rois,
    float* __restrict__ out, int n_rois)
{
  const int n = blockIdx.x;
  if (n >= n_rois) return;
  const int wave = threadIdx.x >> 5;   // 0..15 : channel tile
  const int lane = threadIdx.x & 31;
  const int ct   = wave;

  // ROI scalars (uniform)
  const float rid = rois[n*5+0];
  const float rx0 = rois[n*5+1];
  const float ry0 = rois[n*5+2];
  const float rx1 = rois[n*5+3];
  const float ry1 = rois[n*5+4];
  const int bimg = (int)rid;

  // Replicate reference fp32 arithmetic order exactly.
  const float x0 = rx0 * SSCALE / (float)(FW-1);
  const float y0 = ry0 * SSCALE / (float)(FH-1);
  const float x1 = rx1 * SSCALE / (float)(FW-1);
  const float y1 = ry1 * SSCALE / (float)(FH-1);
  const float dx = (x1 - x0) * (float)(FW-1) / (float)(CROPV-1);
  const float dy = (y1 - y0) * (float)(FH-1) / (float)(CROPV-1);

  // Per-lane x sample for column j = lane&15 (j=14,15 are zero padding).
  const int j = lane & 15;
  const float in_x = x0 * (float)(FW-1) + (float)j * dx;
  const bool  vx   = (in_x >= 0.f) && (in_x <= (float)(FW-1));
  const float left = floorf(in_x);
  const float xw   = in_x - left;
  int xl = (int)left;     xl = xl < 0 ? 0 : (xl > FW-1 ? FW-1 : xl);
  int xr = (int)left + 1; xr = xr < 0 ? 0 : (xr > FW-1 ? FW-1 : xr);
  const float vxf = vx ? 1.f : 0.f;

  // B-matrix chunks. B 4x16 layout: lanes0-15 hold rows {4kc, 4kc+1},
  // lanes16-31 rows {4kc+2, 4kc+3}; column = lane&15 = this lane's own j.
  const int khalf = (lane >> 4) << 1;   // 0 or 2
  v2f Bm[7];
  #pragma unroll
  for (int kc = 0; kc < 7; ++kc) {
    const int kb = 4*kc + khalf;                 // even K row in [0,26]
    const float m = (kb == 2*j) ? vxf : 0.f;     // Wx[2j,j], Wx[2j+1,j]
    Bm[kc].x = m * (1.f - xw);
    Bm[kc].y = m * xw;
  }

  // Columns this lane gathers for its A slots (i-invariant, hoisted).
  // A slot K even -> xl(j2), K odd -> xr(j2), j2 = 2kc + (lane>=16).
  int xlv[7], xrv[7];
  #pragma unroll
  for (int kc = 0; kc < 7; ++kc) {
    const int j2 = 2*kc + (lane >> 4);           // 0..13
    xlv[kc] = __shfl(xl, j2, 32);
    xrv[kc] = __shfl(xr, j2, 32);
  }

  const int cA = ct*16 + (lane & 15);            // A row: channel
  const float* fbase = feat + ((size_t)bimg * NC + cA) * (size_t)(FH*FW);
  float* obase = out + ((size_t)n * NC + (size_t)ct*16) * (CROPV*CROPV);

  for (int i = 0; i < CROPV; ++i) {
    const float in_y = y0 * (float)(FH-1) + (float)i * dy;
    const bool  vy   = (in_y >= 0.f) && (in_y <= (float)(FH-1));
    const float topf = floorf(in_y);
    const float yw   = in_y - topf;
    int yt = (int)topf;     yt = yt < 0 ? 0 : (yt > FH-1 ? FH-1 : yt);
    int yb = (int)topf + 1; yb = yb < 0 ? 0 : (yb > FH-1 ? FH-1 : yb);
    const float vyf = vy ? 1.f : 0.f;
    const float wt = vyf * (1.f - yw);
    const float wb = vyf * yw;
    const float* rowt = fbase + (size_t)yt * FW;
    const float* rowb = fbase + (size_t)yb * FW;

    v8f acc = {0.f,0.f,0.f,0.f,0.f,0.f,0.f,0.f};
    #pragma unroll
    for (int kc = 0; kc < 7; ++kc) {
      const float ftl = rowt[xlv[kc]];
      const float fbl = rowb[xlv[kc]];
      const float ftr = rowt[xrv[kc]];
      const float fbr = rowb[xrv[kc]];
      v2f a;
      a.x = wt*ftl + wb*fbl;   // even K -> xl column, y-blended (+vy mask)
      a.y = wt*ftr + wb*fbr;   // odd  K -> xr column
      acc = __builtin_amdgcn_wmma_f32_16x16x4_f32(
          /*neg_a=*/false, a, /*neg_b=*/false, Bm[kc],
          /*c_mod=*/(short)0, acc, /*reuse_a=*/false, /*reuse_b=*/false);
    }

    // D 16x16: VGPR r -> channel (r + 8*(lane>=16)), col = lane&15.
    const int Nj = lane & 15;
    if (Nj < CROPV) {
      const int Mo = (lane >> 4) * 8;
      float* op = obase + (size_t)i * CROPV + Nj;
      #pragma unroll
      for (int r = 0; r < 8; ++r)
        op[(size_t)(r + Mo) * (CROPV*CROPV)] = acc[r];
    }
  }
}

extern "C" void kernel_launch(void* const* d_in, const int* in_sizes, int n_in,
                              void* d_out, int out_size, void* d_ws, size_t ws_size,
                              hipStream_t stream) {
  const float* feat = (const float*)d_in[0];   // (8, 256, 200, 200) fp32
  const float* rois = (const float*)d_in[1];   // (N, 5) fp32
  float* out = (float*)d_out;                  // (N, 256, 14, 14) fp32
  const int n_rois = in_sizes[1] / 5;          // 1000
  if (n_rois <= 0) return;
  roialign_wmma_kernel<<<dim3((unsigned)n_rois), dim3(512), 0, stream>>>(
      feat, rois, out, n_rois);
}